// ASGCN_HETEROGENEOUS_87814901334298
// MI455X (gfx1250) — compile-verified
//
#include <hip/hip_runtime.h>

// ---------------- problem constants ----------------
#define B_   32
#define S_   512
#define SA_  16
#define E_   256
#define NH_  8
#define HD_  32

typedef float v2f __attribute__((ext_vector_type(2)));
typedef float v8f __attribute__((ext_vector_type(8)));
typedef int v4i_ __attribute__((__vector_size__(4 * sizeof(int))));

// fp32 WMMA: D(16x16) = A(16x4) * B(4x16) + C   (wave32)
// A: lanes 0-15 -> M=lane, K={0,1}; lanes 16-31 -> M=lane-16, K={2,3}
// B: lanes 0-15 -> N=lane, K rows {0,1}; lanes 16-31 -> N=lane-16, K rows {2,3}
// C: lanes 0-15 -> N=lane, M=vgpr(0..7); lanes 16-31 -> N=lane-16, M=vgpr+8
__device__ __forceinline__ v8f wmma4(v2f a, v2f b, v8f c) {
  return __builtin_amdgcn_wmma_f32_16x16x4_f32(false, a, false, b, (short)0, c,
                                               false, false);
}

// pointer casts for the async global->LDS builtin (signature leaked by clang:
// first param is '__vector_size__(16) int __device__ *')
#define GV4(p) ((__attribute__((address_space(1))) v4i_*)(p))
#define LV4(p) ((__attribute__((address_space(3))) v4i_*)(p))

// ---------------- tiny kernels ----------------

// lengths: text_len[b] = count(text!=0)+5 ; aspect_len[b] = count(aspect!=0)
__global__ __launch_bounds__(256) void k_len(const int* __restrict__ ti,
                                             const int* __restrict__ ai,
                                             float* __restrict__ tlen,
                                             float* __restrict__ alen) {
  __shared__ int c1, c2;
  int b = blockIdx.x, tid = threadIdx.x;
  if (tid == 0) { c1 = 0; c2 = 0; }
  __syncthreads();
  int lc = 0;
  for (int i = tid; i < S_; i += 256) lc += (ti[b * S_ + i] != 0);
  int la = (tid < SA_) ? (ai[b * SA_ + tid] != 0) : 0;
  if (lc) atomicAdd(&c1, lc);
  if (la) atomicAdd(&c2, la);
  __syncthreads();
  if (tid == 0) {
    tlen[b] = (float)c1 + 5.0f;
    alen[b] = (float)c2;
  }
}

// embedding gather: out[row][e] = embed[idx[row]][e]; grid = rows, block = 256
__global__ __launch_bounds__(256) void k_gather(const int* __restrict__ idx,
                                                const float* __restrict__ embed,
                                                float* __restrict__ out) {
  size_t row = blockIdx.x;
  int e = threadIdx.x;
  out[row * E_ + e] = embed[(size_t)idx[row] * E_ + e];
}

// per-head score projection: o[(b*8+h)*Ls + s] = dot(X[(b*Ls+s)*256 + h*32 .. +31], w)
__global__ __launch_bounds__(256) void k_skq(const float* __restrict__ X,
                                             const float* __restrict__ w,
                                             float* __restrict__ o, int Ls) {
  int g = blockIdx.x * 256 + threadIdx.x;  // (b*8+h)*Ls + s
  int s = g % Ls;
  int bh = g / Ls;
  int b = bh >> 3, h = bh & 7;
  const float* r = &X[((size_t)(b * Ls + s)) * E_ + h * HD_];
  float acc = 0.f;
#pragma unroll
  for (int d = 0; d < HD_; ++d) acc += r[d] * w[d];
  o[g] = acc;
}

// row dots for GAT: as[row]=dot(gh[row],att_src), ad[row]=dot(gh[row],att_dst)
__global__ __launch_bounds__(256) void k_srcdst(const float* __restrict__ gh,
                                                const float* __restrict__ ws_,
                                                const float* __restrict__ wd_,
                                                float* __restrict__ as_,
                                                float* __restrict__ ad_) {
  int wave = threadIdx.x >> 5, lane = threadIdx.x & 31;
  size_t row = (size_t)blockIdx.x * 8 + wave;
  const float* r = &gh[row * E_];
  float s = 0.f, d = 0.f;
  for (int i = lane; i < E_; i += 32) {
    float x = r[i];
    s += x * ws_[i];
    d += x * wd_[i];
  }
  for (int m = 16; m >= 1; m >>= 1) {
    s += __shfl_xor(s, m);
    d += __shfl_xor(d, m);
  }
  if (lane == 0) { as_[row] = s; ad_[row] = d; }
}

// mean over sequence: out[b][e] = sum_s in[b][s][e] / len[b]
__global__ __launch_bounds__(256) void k_mean(const float* __restrict__ in,
                                              const float* __restrict__ len,
                                              float* __restrict__ out, int Ls) {
  int b = blockIdx.x, e = threadIdx.x;
  float s = 0.f;
  for (int i = 0; i < Ls; ++i) s += in[((size_t)b * Ls + i) * E_ + e];
  out[b * E_ + e] = s / len[b];
}

// final FC: out[b][p] = fcW[p] . [x_mean, h_text_mean, h_aspect_mean] + fcb[p]
__global__ __launch_bounds__(128) void k_fc(const float* __restrict__ xm,
                                            const float* __restrict__ tm,
                                            const float* __restrict__ am,
                                            const float* __restrict__ fcW,
                                            const float* __restrict__ fcb,
                                            float* __restrict__ out) {
  int t = threadIdx.x;
  if (t >= B_ * 3) return;
  int b = t / 3, p = t % 3;
  const float* w = &fcW[p * 3 * E_];
  float s = fcb[p];
  for (int i = 0; i < E_; ++i) {
    s += w[i] * xm[b * E_ + i];
    s += w[E_ + i] * tm[b * E_ + i];
    s += w[2 * E_ + i] * am[b * E_ + i];
  }
  out[b * 3 + p] = s;
}

// ---------------- WMMA GEMM: Y[M,256] = act(X[M,256] @ W[256,256]^T + bias) ----
// grid.x = M/16; block = 256 (8 waves); wave w covers N in [32w, 32w+32)
__global__ __launch_bounds__(256) void k_gemm(const float* __restrict__ X,
                                              const float* __restrict__ W,
                                              const float* __restrict__ bias,
                                              float* __restrict__ Y, int relu) {
  __shared__ float sA[16][E_ + 4];  // row stride 260 -> conflict-free A reads
  int tid = threadIdx.x;
  size_t row0 = (size_t)blockIdx.x * 16;
#if defined(__gfx1250__) && __has_builtin(__builtin_amdgcn_global_load_async_to_lds_b128)
  // async global->LDS staging of the 16x256 A tile (1024 x 16B chunks)
  for (int i = tid; i < 16 * (E_ / 4); i += 256) {
    int r = i >> 6;          // 0..15
    int c = (i & 63) << 2;   // 0..252 step 4 (16B aligned both sides)
    __builtin_amdgcn_global_load_async_to_lds_b128(
        GV4(&X[(row0 + r) * E_ + c]), LV4(&sA[r][c]), 0, 0);
  }
#if __has_builtin(__builtin_amdgcn_s_wait_asynccnt)
  __builtin_amdgcn_s_wait_asynccnt(0);
#else
  asm volatile("s_wait_asynccnt 0" ::: "memory");
#endif
  __syncthreads();
#else
  for (int i = tid; i < 16 * E_; i += 256) {
    int r = i >> 8, c = i & 255;
    sA[r][c] = X[(row0 + r) * E_ + c];
  }
  __syncthreads();
#endif
  int wave = tid >> 5, lane = tid & 31;
  int nbase = wave * 32;
  int lhalf = lane >> 4, lidx = lane & 15;
  v8f acc0 = {}, acc1 = {};
  for (int k = 0; k < E_; k += 4) {
    int ka = k + 2 * lhalf;
    v2f a;
    a.x = sA[lidx][ka];
    a.y = sA[lidx][ka + 1];
    const float* w0 = &W[(size_t)(nbase + lidx) * E_ + ka];
    const float* w1 = &W[(size_t)(nbase + 16 + lidx) * E_ + ka];
    v2f b0, b1;
    b0.x = w0[0]; b0.y = w0[1];
    b1.x = w1[0]; b1.y = w1[1];
    acc0 = wmma4(a, b0, acc0);
    acc1 = wmma4(a, b1, acc1);
  }
#pragma unroll
  for (int i = 0; i < 8; ++i) {
    int m = i + 8 * lhalf;
    int n0 = nbase + lidx, n1 = nbase + 16 + lidx;
    float v0 = acc0[i] + (bias ? bias[n0] : 0.f);
    float v1 = acc1[i] + (bias ? bias[n1] : 0.f);
    if (relu) { v0 = fmaxf(v0, 0.f); v1 = fmaxf(v1, 0.f); }
    Y[(row0 + m) * E_ + n0] = v0;
    Y[(row0 + m) * E_ + n1] = v1;
  }
}

// ------- fused MLP-attention: tanh+softmax -> stream score -> WMMA score@kx ----
// grid = B*NH*(Lq/16); block = 256
__global__ __launch_bounds__(256) void k_attn(const float* __restrict__ kx,
                                              const float* __restrict__ sk,
                                              const float* __restrict__ sq,
                                              float* __restrict__ score_out,
                                              float* __restrict__ att_out,
                                              int Lq) {
  __shared__ float sc[16][S_ + 4];        // padded: conflict-free WMMA A reads
  __shared__ float redC[8][2][8][32];     // 16 KB: per-wave partial C
  int qtiles = Lq >> 4;
  int bid = blockIdx.x;
  int qt = bid % qtiles;
  int h = (bid / qtiles) % NH_;
  int b = bid / (qtiles * NH_);
  int q0 = qt * 16;
  int tid = threadIdx.x;
  int row = tid >> 4;     // q-row in tile (16 threads per row, one half-wave)
  int jg = tid & 15;

  const float* skr = &sk[(size_t)(b * NH_ + h) * S_];
  float sqv = sq[(size_t)(b * NH_ + h) * Lq + q0 + row];

  float lmax = -INFINITY;
  for (int j = jg; j < S_; j += 16) {
    float e = tanhf(sqv + skr[j]);
    sc[row][j] = e;
    lmax = fmaxf(lmax, e);
  }
  for (int m = 8; m >= 1; m >>= 1) lmax = fmaxf(lmax, __shfl_xor(lmax, m));
  float lsum = 0.f;
  for (int j = jg; j < S_; j += 16) {
    float ev = __expf(sc[row][j] - lmax);
    sc[row][j] = ev;
    lsum += ev;
  }
  for (int m = 8; m >= 1; m >>= 1) lsum += __shfl_xor(lsum, m);
  float inv = 1.f / lsum;
  float* so = &score_out[((size_t)(h * B_ + b) * Lq + q0 + row) * S_];
  for (int j = jg; j < S_; j += 16) {
    float wv = sc[row][j] * inv;
    sc[row][j] = wv;
    __builtin_nontemporal_store(wv, &so[j]);  // 268MB stream: bypass L2 retention
  }
  __syncthreads();

  // out(16 x 32) = sc(16 x 512) @ kx[b,:,h,:](512 x 32); K split across 8 waves
  int wave = tid >> 5, lane = tid & 31;
  int lhalf = lane >> 4, lidx = lane & 15;
  const float* kxb = &kx[(size_t)b * S_ * E_ + h * HD_];
  v8f acc0 = {}, acc1 = {};
  for (int k = wave * 64; k < wave * 64 + 64; k += 4) {
    int ka = k + 2 * lhalf;
    v2f a;
    a.x = sc[lidx][ka];
    a.y = sc[lidx][ka + 1];
    const float* bp = &kxb[(size_t)ka * E_ + lidx];
    v2f b0, b1;
    b0.x = bp[0];  b0.y = bp[E_];
    b1.x = bp[16]; b1.y = bp[16 + E_];
    acc0 = wmma4(a, b0, acc0);
    acc1 = wmma4(a, b1, acc1);
  }
#pragma unroll
  for (int i = 0; i < 8; ++i) {
    redC[wave][0][i][lane] = acc0[i];
    redC[wave][1][i][lane] = acc1[i];
  }
  __syncthreads();
  for (int o = tid; o < 512; o += 256) {  // 16 q x 32 d outputs
    int m = o >> 5, d = o & 31;
    int ah = d >> 4;
    int ln = (d & 15) + 16 * (m >> 3);
    int vg = m & 7;
    float s = 0.f;
#pragma unroll
    for (int w2 = 0; w2 < 8; ++w2) s += redC[w2][ah][vg][ln];
    att_out[((size_t)b * Lq + q0 + m) * E_ + h * HD_ + d] = s;
  }
}

// ------- fused GAT layer: masked softmax over sources -> WMMA w^T @ h --------
// grid = B*(S/16); block = 256. out = relu(einsum('ji,jc->ic', w, h) + bias)
__global__ __launch_bounds__(256) void k_gat(const float* __restrict__ gh,
                                             const float* __restrict__ asrc,
                                             const float* __restrict__ adst,
                                             const float* __restrict__ adj,
                                             const float* __restrict__ bias,
                                             float* __restrict__ out) {
  __shared__ float sc[16][S_ + 4];  // w[j, i0+row] stored as sc[row][j], padded
  int bid = blockIdx.x;
  int it = bid & 31, b = bid >> 5;
  int i0 = it * 16;
  int tid = threadIdx.x;
  int row = tid >> 4;  // target index within tile
  int jg = tid & 15;

  float adv = adst[b * S_ + i0 + row];
  const float* asb = &asrc[b * S_];
  const float* adjb = &adj[(size_t)b * S_ * S_];

  float lmax = -INFINITY;
  for (int j = jg; j < S_; j += 16) {
    float e = asb[j] + adv;
    e = e > 0.f ? e : 0.2f * e;  // leaky_relu(0.2)
    float av = __builtin_nontemporal_load(&adjb[(size_t)j * S_ + i0 + row]);
    bool msk = av > 0.f;
    sc[row][j] = msk ? e : -INFINITY;
    if (msk) lmax = fmaxf(lmax, e);
  }
  for (int m = 8; m >= 1; m >>= 1) lmax = fmaxf(lmax, __shfl_xor(lmax, m));
  if (lmax == -INFINITY) lmax = 0.f;  // isolated target
  float lsum = 0.f;
  for (int j = jg; j < S_; j += 16) {
    float v = sc[row][j];
    float ev = (v == -INFINITY) ? 0.f : __expf(v - lmax);
    sc[row][j] = ev;
    lsum += ev;
  }
  for (int m = 8; m >= 1; m >>= 1) lsum += __shfl_xor(lsum, m);
  float inv = 1.f / fmaxf(lsum, 1e-16f);
  for (int j = jg; j < S_; j += 16) sc[row][j] *= inv;
  __syncthreads();

  // out(16 x 256) = sc(16 x 512) @ gh[b](512 x 256); wave owns 32-wide C slice
  int wave = tid >> 5, lane = tid & 31;
  int lhalf = lane >> 4, lidx = lane & 15;
  int cbase = wave * 32;
  const float* ghb = &gh[(size_t)b * S_ * E_];
  v8f acc0 = {}, acc1 = {};
  for (int k = 0; k < S_; k += 4) {
    int ka = k + 2 * lhalf;
    v2f a;
    a.x = sc[lidx][ka];
    a.y = sc[lidx][ka + 1];
    const float* bp = &ghb[(size_t)ka * E_ + cbase + lidx];
    v2f b0, b1;
    b0.x = bp[0];  b0.y = bp[E_];
    b1.x = bp[16]; b1.y = bp[16 + E_];
    acc0 = wmma4(a, b0, acc0);
    acc1 = wmma4(a, b1, acc1);
  }
#pragma unroll
  for (int i = 0; i < 8; ++i) {
    int m = i + 8 * lhalf;
    int c0 = cbase + lidx, c1 = cbase + 16 + lidx;
    out[((size_t)b * S_ + i0 + m) * E_ + c0] = fmaxf(acc0[i] + bias[c0], 0.f);
    out[((size_t)b * S_ + i0 + m) * E_ + c1] = fmaxf(acc1[i] + bias[c1], 0.f);
  }
}

// ---------------- host-side orchestration ----------------
extern "C" void kernel_launch(void* const* d_in, const int* in_sizes, int n_in,
                              void* d_out, int out_size, void* d_ws,
                              size_t ws_size, hipStream_t stream) {
  (void)in_sizes; (void)n_in; (void)out_size; (void)ws_size;
  // inputs in setup_inputs() insertion order (params nested depth-first)
  const int* ti = (const int*)d_in[0];      // text_indices (B,S)
  const int* ai = (const int*)d_in[1];      // aspect_indices (B,SA)
  /* d_in[2] = left_indices: unused by reference */
  const float* adj = (const float*)d_in[3]; // (B,S,S)
  const float* embed = (const float*)d_in[4];
  auto F = [&](int i) { return (const float*)d_in[i]; };
  // attn_k: 5..11   attn_q: 12..18   ffn_c: 19..22   ffn_t: 23..26
  // gc1: 27..30     gc2: 31..34      fcW: 35  fcb: 36
  const float *k_wkW = F(5), *k_wkb = F(6), *k_wqW = F(7), *k_wqb = F(8),
              *k_prW = F(9), *k_prb = F(10), *k_wm = F(11);
  const float *q_wkW = F(12), *q_wkb = F(13), *q_wqW = F(14), *q_wqb = F(15),
              *q_prW = F(16), *q_prb = F(17), *q_wm = F(18);
  const float *c_w1W = F(19), *c_w1b = F(20), *c_w2W = F(21), *c_w2b = F(22);
  const float *t_w1W = F(23), *t_w1b = F(24), *t_w2W = F(25), *t_w2b = F(26);
  const float *g1W = F(27), *g1s = F(28), *g1d = F(29), *g1b = F(30);
  const float *g2W = F(31), *g2s = F(32), *g2d = F(33), *g2b = F(34);
  const float *fcW = F(35), *fcb = F(36);

  float* W = (float*)d_ws;
  float* TEXT = W + 0;             // B*S*E      = 4194304
  float* ASPE = W + 4194304;       // B*SA*E     = 131072
  float* KX   = W + 4325376;       // 4194304
  float* QX   = W + 8519680;       // 4194304
  float* SK   = W + 12713984;      // B*NH*S     = 131072
  float* SQ   = W + 12845056;      // 131072
  float* ATT  = W + 12976128;      // 4194304
  float* H1   = W + 17170432;      // 4194304
  float* H2   = W + 21364736;      // 4194304 (h_text)
  float* GH   = W + 25559040;      // 4194304
  float* GX   = W + 29753344;      // 4194304
  float* AS   = W + 33947648;      // 16384
  float* AD   = W + 33964032;      // 16384
  float* TM   = W + 33980416;      // 8192
  float* AM   = W + 33988608;      // 8192
  float* XM   = W + 33996800;      // 8192
  float* TLEN = W + 34004992;      // 32
  float* ALEN = W + 34005024;      // 32
  float* AATT = W + 34005056;      // B*SA*E = 131072
  float* AH1  = W + 34136128;      // 131072
  float* AH2  = W + 34267200;      // 131072 (h_aspect)

  float* out = (float*)d_out;
  float* out_logits = out;                       // (B,3)
  float* out_tscore = out + 96;                  // (NH*B, S, S)
  float* out_ascore = out + 96 + (size_t)NH_ * B_ * S_ * S_;  // (NH*B, SA, S)

  const int Mtxt = B_ * S_, Masp = B_ * SA_;

  k_len<<<B_, 256, 0, stream>>>(ti, ai, TLEN, ALEN);
  k_gather<<<Mtxt, 256, 0, stream>>>(ti, embed, TEXT);
  k_gather<<<Masp, 256, 0, stream>>>(ai, embed, ASPE);

  // ---- text self-attention (attn_k) ----
  k_gemm<<<Mtxt / 16, 256, 0, stream>>>(TEXT, k_wkW, k_wkb, KX, 0);
  k_gemm<<<Mtxt / 16, 256, 0, stream>>>(TEXT, k_wqW, k_wqb, QX, 0);
  k_skq<<<(B_ * NH_ * S_) / 256, 256, 0, stream>>>(KX, k_wm, SK, S_);
  k_skq<<<(B_ * NH_ * S_) / 256, 256, 0, stream>>>(QX, k_wm + HD_, SQ, S_);
  k_attn<<<B_ * NH_ * (S_ / 16), 256, 0, stream>>>(KX, SK, SQ, out_tscore, ATT, S_);
  k_gemm<<<Mtxt / 16, 256, 0, stream>>>(ATT, k_prW, k_prb, H1, 0);
  // ffn_c
  k_gemm<<<Mtxt / 16, 256, 0, stream>>>(H1, c_w1W, c_w1b, QX, 1);   // reuse QX
  k_gemm<<<Mtxt / 16, 256, 0, stream>>>(QX, c_w2W, c_w2b, H2, 0);   // h_text
  k_mean<<<B_, 256, 0, stream>>>(H2, TLEN, TM, S_);

  // ---- aspect attention (attn_q): k = text, q = aspect ----
  k_gemm<<<Mtxt / 16, 256, 0, stream>>>(TEXT, q_wkW, q_wkb, KX, 0);
  k_gemm<<<Masp / 16, 256, 0, stream>>>(ASPE, q_wqW, q_wqb, QX, 0);
  k_skq<<<(B_ * NH_ * S_) / 256, 256, 0, stream>>>(KX, q_wm, SK, S_);
  k_skq<<<(B_ * NH_ * SA_) / 256, 256, 0, stream>>>(QX, q_wm + HD_, SQ, SA_);
  k_attn<<<B_ * NH_, 256, 0, stream>>>(KX, SK, SQ, out_ascore, AATT, SA_);
  k_gemm<<<Masp / 16, 256, 0, stream>>>(AATT, q_prW, q_prb, AH1, 0);
  // ffn_t
  k_gemm<<<Masp / 16, 256, 0, stream>>>(AH1, t_w1W, t_w1b, AATT, 1);
  k_gemm<<<Masp / 16, 256, 0, stream>>>(AATT, t_w2W, t_w2b, AH2, 0);
  k_mean<<<B_, 256, 0, stream>>>(AH2, ALEN, AM, SA_);

  // ---- GAT layer 1 on h_text ----
  k_gemm<<<Mtxt / 16, 256, 0, stream>>>(H2, g1W, nullptr, GH, 0);
  k_srcdst<<<Mtxt / 8, 256, 0, stream>>>(GH, g1s, g1d, AS, AD);
  k_gat<<<B_ * (S_ / 16), 256, 0, stream>>>(GH, AS, AD, adj, g1b, GX);
  // ---- GAT layer 2 ----
  k_gemm<<<Mtxt / 16, 256, 0, stream>>>(GX, g2W, nullptr, GH, 0);
  k_srcdst<<<Mtxt / 8, 256, 0, stream>>>(GH, g2s, g2d, AS, AD);
  k_gat<<<B_ * (S_ / 16), 256, 0, stream>>>(GH, AS, AD, adj, g2b, ATT);  // final x

  k_mean<<<B_, 256, 0, stream>>>(ATT, TLEN, XM, S_);
  k_fc<<<1, 128, 0, stream>>>(XM, TM, AM, fcW, fcb, out_logits);
}